// QOuterC_15685220565666
// MI455X (gfx1250) — compile-verified
//
#include <hip/hip_runtime.h>

// out[part, t, b, p, q]:
//   real: r_p r_q + i_p i_q   -> A=[r|i] (16x4, K=0,1 used), B=[r|i] (4x16)
//   imag: i_p r_q - r_p i_q   -> A=[i|-r], same B
// One wave handles one (t,b) slab: 4x4 tiles of 16x16, two WMMAs per tile.

typedef __attribute__((ext_vector_type(2))) float v2f;
typedef __attribute__((ext_vector_type(8))) float v8f;

#define BB 32
#define TT 256
#define DD 64

__global__ __launch_bounds__(256) void qouterc_wmma_kernel(
    const float* __restrict__ re, const float* __restrict__ im,
    float* __restrict__ out) {
  const int lane = threadIdx.x & 31;
  const int waveInBlock = threadIdx.x >> 5;
  const int slab = blockIdx.x * 8 + waveInBlock;  // slab = t*B + b, 8192 total
  const int t = slab >> 5;   // / B
  const int b = slab & 31;   // % B

  // Inputs are [B, T, D]
  const float* rp = re + ((size_t)b * TT + t) * DD;
  const float* ip = im + ((size_t)b * TT + t) * DD;

  const bool lo = (lane < 16);     // lanes 0-15 carry K=0,1; lanes 16-31 carry K=2,3 (zeros)
  const int m = lane & 15;

  // Preload the 64-element r and i vectors as four 16-element segments.
  float rv[4], iv[4];
#pragma unroll
  for (int s = 0; s < 4; ++s) {
    float rx = rp[s * 16 + m];
    float ix = ip[s * 16 + m];
    rv[s] = lo ? rx : 0.0f;
    iv[s] = lo ? ix : 0.0f;
  }

  // Output base pointers: out[2, T, B, D, D]
  float* oR = out + ((size_t)t * BB + b) * (DD * DD);
  float* oI = oR + (size_t)TT * BB * DD * DD;

#pragma unroll
  for (int tp = 0; tp < 4; ++tp) {
    v2f aR;  aR.x = rv[tp];  aR.y = iv[tp];    // A = [r | i]
    v2f aI;  aI.x = iv[tp];  aI.y = -rv[tp];   // A' = [i | -r]
    const int rowBase = tp * 16 + (lo ? 0 : 8);
#pragma unroll
    for (int tq = 0; tq < 4; ++tq) {
      v2f bR;  bR.x = rv[tq];  bR.y = iv[tq];  // B = [r | i] (4x16)
      v8f cR = {};
      v8f cI = {};
      // D = A x B + 0 : 16x16 f32 tile per instruction
      cR = __builtin_amdgcn_wmma_f32_16x16x4_f32(
          /*neg_a=*/false, aR, /*neg_b=*/false, bR,
          /*c_mod=*/(short)0, cR, /*reuse_a=*/false, /*reuse_b=*/false);
      cI = __builtin_amdgcn_wmma_f32_16x16x4_f32(
          /*neg_a=*/false, aI, /*neg_b=*/false, bR,
          /*c_mod=*/(short)0, cI, /*reuse_a=*/false, /*reuse_b=*/false);

      const int col = tq * 16 + m;
#pragma unroll
      for (int v = 0; v < 8; ++v) {
        const int idx = (rowBase + v) * DD + col;
        oR[idx] = cR[v];
        oI[idx] = cI[v];
      }
    }
  }
}

extern "C" void kernel_launch(void* const* d_in, const int* in_sizes, int n_in,
                              void* d_out, int out_size, void* d_ws, size_t ws_size,
                              hipStream_t stream) {
  (void)in_sizes; (void)n_in; (void)out_size; (void)d_ws; (void)ws_size;
  const float* re = (const float*)d_in[0];
  const float* im = (const float*)d_in[1];
  float* out = (float*)d_out;
  // 8192 waves total: 1024 blocks x 256 threads (8 waves/block), exact coverage
  qouterc_wmma_kernel<<<dim3(1024), dim3(256), 0, stream>>>(re, im, out);
}